// STD_Hunter_71004399337581
// MI455X (gfx1250) — compile-verified
//
#include <hip/hip_runtime.h>
#include <hip/hip_bf16.h>
#include <math.h>

#define NN   2000
#define BB   16
#define CC   32
#define TT   12
#define LL   3
#define RK   4
#define CNq  8
#define TP1  13          // T+1 time columns in raw input
#define KPAD 8           // padded taps per input channel (taps 0..6 used, 7 zero)
#define KTOT (CC*KPAD)   // 256 = GEMM K for the conv
#define NCOL (BB*TT)     // 192 = GEMM N (all batch*time columns)
#define XEXT 20          // circularly extended time axis (t+j <= 18)
#define D_CW 13824       // CN*C*18*L
#define D_CB 96          // CN*4*L
#define D_RW 3072        // C*C*L
#define D_RB 96          // C*L
#define XEL  (BB*CC*NN*TT)   // 12,288,000
#define SKEL (BB*CC*NN)      // 1,024,000

typedef float vf2 __attribute__((ext_vector_type(2)));
typedef float vf8 __attribute__((ext_vector_type(8)));

__device__ __forceinline__ vf8 wmma_f32x4(vf2 a, vf2 b, vf8 c) {
  // D = A(16x4,f32) * B(4x16,f32) + C(16x16,f32)   -> v_wmma_f32_16x16x4_f32
  return __builtin_amdgcn_wmma_f32_16x16x4_f32(false, a, false, b, (short)0, c, false, false);
}

// Branchless hardware transcendentals: keeps EXEC all-ones between WMMA regions.
__device__ __forceinline__ float fast_tanh(float x) {
#if defined(__has_builtin) && __has_builtin(__builtin_amdgcn_tanhf)
  return __builtin_amdgcn_tanhf(x);            // v_tanh_f32
#else
  float r;
  asm volatile("v_tanh_f32 %0, %1" : "=v"(r) : "v"(x));
  return r;
#endif
}
__device__ __forceinline__ float fast_sigmoid(float x) {
  // v_exp_f32 is exp2 in hardware, full range: overflow->inf->rcp 0, underflow->0->1
  float e = __builtin_amdgcn_exp2f(-x * 1.44269504088896340736f);
  return __builtin_amdgcn_rcpf(1.0f + e);      // v_exp_f32 + v_add + v_rcp_f32
}

// ---------------------------------------------------------------- month/char
__global__ void k_charm(const float* __restrict__ inp, const float* __restrict__ chars,
                        float* __restrict__ charm) {
  int n = blockIdx.x * 256 + threadIdx.x;
  if (n >= NN) return;
  int m = ((int)inp[TT]) % 12; if (m < 0) m += 12;   // input[0,0,0,-1]
  int mp = (m + 11) % 12, mn = (m + 1) % 12;
  for (int r = 0; r < RK; ++r) {
    float v = chars[(n*12 + mp)*RK + r] + chars[(n*12 + m)*RK + r] + chars[(n*12 + mn)*RK + r];
    charm[n*RK + r] = v * (1.0f/3.0f);
  }
}

// ---------------------------------------------------------------- ch = adj @ char_m
__global__ __launch_bounds__(128) void k_ch(const float* __restrict__ adj,
                                            const float* __restrict__ charm,
                                            float* __restrict__ ch) {
  __shared__ float red[128*4];
  int n = blockIdx.x, tid = threadIdx.x;
  float a0=0.f, a1=0.f, a2=0.f, a3=0.f;
  for (int m = tid; m < NN; m += 128) {
    float a = adj[n*NN + m];
    a0 += a*charm[m*4+0]; a1 += a*charm[m*4+1];
    a2 += a*charm[m*4+2]; a3 += a*charm[m*4+3];
  }
  red[tid*4+0]=a0; red[tid*4+1]=a1; red[tid*4+2]=a2; red[tid*4+3]=a3;
  __syncthreads();
  for (int s = 64; s > 0; s >>= 1) {
    if (tid < s)
      for (int r = 0; r < 4; ++r) red[tid*4+r] += red[(tid+s)*4+r];
    __syncthreads();
  }
  if (tid < 4) ch[n*4 + tid] = red[tid];
}

// ---------------------------------------------------------------- start conv x0
__global__ __launch_bounds__(256) void k_init(const float* __restrict__ inp,
                                              const float* __restrict__ stw,
                                              const float* __restrict__ stb,
                                              float* __restrict__ X) {
  int idx = blockIdx.x * 256 + threadIdx.x;          // grid sized exactly XEL
  int t = idx % TT; int r1 = idx / TT;
  int n = r1 % NN;  int r2 = r1 / NN;
  int c = r2 % CC;  int b = r2 / CC;
  float i0 = inp[((b*2 + 0)*NN + n)*TP1 + t];
  float i1 = inp[((b*2 + 1)*NN + n)*TP1 + t];
  X[idx] = i0*stw[c*2+0] + i1*stw[c*2+1] + stb[c];
}

// ---------------------------------------------------------------- skip0
__global__ __launch_bounds__(256) void k_skip0(const float* __restrict__ inp,
                                               const float* __restrict__ s0w,
                                               const float* __restrict__ s0b,
                                               float* __restrict__ skip) {
  int idx = blockIdx.x * 256 + threadIdx.x;          // grid sized exactly SKEL
  int n = idx % NN; int r1 = idx / NN;
  int c = r1 % CC;  int b = r1 / CC;
  float acc = s0b[c];
  for (int i = 0; i < 2; ++i)
    for (int t = 0; t < TT; ++t)
      acc += inp[((b*2 + i)*NN + n)*TP1 + t] * s0w[(c*2 + i)*TT + t];
  skip[idx] = acc;
}

// ---------------------------------------------------------------- gated conv (WMMA)
__global__ __launch_bounds__(256) void k_conv(const float* __restrict__ ch,
    const float* __restrict__ Wcw, const float* __restrict__ bcw,
    const float* __restrict__ Wcb, const float* __restrict__ bcb,
    const float* __restrict__ Wgw, const float* __restrict__ bgw,
    const float* __restrict__ Wgb, const float* __restrict__ bgb,
    const float* __restrict__ X, float* __restrict__ Y, int layer) {
  extern __shared__ float sm[];
  float* sWf = sm;                    // [32][256] filt weights, tap-padded
  float* sWg = sm + CC*KTOT;          // [32][256] gate weights
  float* sX  = sm + 2*CC*KTOT;        // [32][16][20] circularly extended x
  float* sBf = sm + 2*CC*KTOT + CC*BB*XEXT;        // [32]
  float* sBg = sBf + CC;                            // [32]
  const int n = blockIdx.x, tid = threadIdx.x;
  const float c0v = ch[n*4+0], c1v = ch[n*4+1], c2v = ch[n*4+2], c3v = ch[n*4+3];

  // build per-node zero-padded weights: k = c*8 + j; slice s = o>>3
  for (int idx = tid; idx < CC*KTOT; idx += 256) {
    int o = idx >> 8, k = idx & 255, c = k >> 3, j = k & 7;
    int s = o >> 3;
    int lo = (0x0B050200u >> (s*8)) & 0xFF;   // {0,2,5,11}
    int kw = (0x07060302u >> (s*8)) & 0xFF;   // {2,3,6,7}
    float vf = 0.f, vg = 0.f;
    if (j < kw) {
      int col = layer*4608 + ((o & 7)*CC + c)*18 + lo + j;
      vf = c0v*Wcw[col] + c1v*Wcw[D_CW+col] + c2v*Wcw[2*D_CW+col] + c3v*Wcw[3*D_CW+col] + bcw[col];
      vg = c0v*Wgw[col] + c1v*Wgw[D_CW+col] + c2v*Wgw[2*D_CW+col] + c3v*Wgw[3*D_CW+col] + bgw[col];
    }
    sWf[idx] = vf; sWg[idx] = vg;
  }
  // biases: reproduce the reference's flat (N,32)->slices reshape scramble
  if (tid < 64) {
    int ty = tid >> 5, o = tid & 31, s = o >> 3, ol = o & 7;
    int flatpos = s*NN*CNq + n*CNq + ol;
    int n2 = flatpos >> 5, jj = flatpos & 31, colW = layer*32 + jj;
    const float* Wb = ty ? Wgb : Wcb;
    const float* bb = ty ? bgb : bcb;
    float v = ch[n2*4+0]*Wb[colW] + ch[n2*4+1]*Wb[D_CB+colW]
            + ch[n2*4+2]*Wb[2*D_CB+colW] + ch[n2*4+3]*Wb[3*D_CB+colW] + bb[colW];
    (ty ? sBg : sBf)[o] = v;
  }
  // x tile with circular time extension (t' in [0,20), value x[..., t'%12])
  for (int idx = tid; idx < CC*BB*XEXT; idx += 256) {
    int c = idx / (BB*XEXT), rem = idx % (BB*XEXT), b = rem / XEXT, tp = rem % XEXT;
    int t = (tp >= TT) ? tp - TT : tp;
    sX[idx] = X[((b*CC + c)*NN + n)*TT + t];
  }
  __syncthreads();

  const int wave = tid >> 5, lane = tid & 31, half = lane >> 4, lr = lane & 15;
  for (int pt = wave; pt < 24; pt += 8) {          // 24 output tiles, 8 waves
    int mt = pt / 12, nt = pt % 12;
    int Mbase = mt*16, Nbase = nt*16;
    int ncol = Nbase + lr, bcol = ncol/12, tcol = ncol%12;
    int xbase = bcol*XEXT + tcol;
    int arow  = (Mbase + lr) * KTOT;
    vf8 accf, accg;
#pragma unroll
    for (int r = 0; r < 8; ++r) {
      int m = Mbase + r + 8*half;
      accf[r] = sBf[m]; accg[r] = sBg[m];
    }
#pragma unroll 4
    for (int k0 = 0; k0 < KTOT; k0 += 4) {
      int kk = k0 + 2*half;
      vf2 af, ag, bx;
      af.x = sWf[arow + kk]; af.y = sWf[arow + kk + 1];
      ag.x = sWg[arow + kk]; ag.y = sWg[arow + kk + 1];
      int ca = kk >> 3,      ja = kk & 7;
      int cb = (kk+1) >> 3,  jb = (kk+1) & 7;
      bx.x = sX[ca*(BB*XEXT) + xbase + ja];
      bx.y = sX[cb*(BB*XEXT) + xbase + jb];
      accf = wmma_f32x4(af, bx, accf);
      accg = wmma_f32x4(ag, bx, accg);
    }
#pragma unroll
    for (int r = 0; r < 8; ++r) {
      int m = Mbase + r + 8*half;
      float fv = fast_tanh(accf[r]);
      float gv = fast_sigmoid(accg[r]);
      Y[((bcol*CC + m)*NN + n)*TT + tcol] = fv * gv;
    }
  }
}

// ------------------------------------------- residual mix + skip + LN partials (WMMA)
__global__ __launch_bounds__(256) void k_resid(const float* __restrict__ ch,
    const float* __restrict__ Wrw, const float* __restrict__ brw,
    const float* __restrict__ Wrb, const float* __restrict__ brb,
    const float* __restrict__ skw, const float* __restrict__ skb,
    const float* __restrict__ X, float* __restrict__ Y,
    float* __restrict__ skip, float* __restrict__ part, int layer) {
  __shared__ float sY[CC*NCOL];   // gated x tile [c][b*12+t]
  __shared__ float sW[CC*CC];     // rw[o][c]
  __shared__ float srb[CC];
  __shared__ float sPS[3*256];    // per-thread per-iter partial sums
  __shared__ float sPQ[3*256];
  const int n = blockIdx.x, tid = threadIdx.x;
  const float c0v = ch[n*4+0], c1v = ch[n*4+1], c2v = ch[n*4+2], c3v = ch[n*4+3];

  for (int idx = tid; idx < CC*CC; idx += 256) {
    int col = layer*1024 + idx;
    sW[idx] = c0v*Wrw[col] + c1v*Wrw[D_RW+col] + c2v*Wrw[2*D_RW+col] + c3v*Wrw[3*D_RW+col] + brw[col];
  }
  if (tid < CC) {
    int col = layer*32 + tid;
    srb[tid] = c0v*Wrb[col] + c1v*Wrb[D_RB+col] + c2v*Wrb[2*D_RB+col] + c3v*Wrb[3*D_RB+col] + brb[col];
  }
  for (int idx = tid; idx < CC*NCOL; idx += 256) {
    int c = idx / NCOL, col = idx % NCOL, b = col/12, t = col%12;
    sY[idx] = Y[((b*CC + c)*NN + n)*TT + t];
  }
  __syncthreads();

  const int wave = tid >> 5, lane = tid & 31, half = lane >> 4, lr = lane & 15;
  for (int it = 0; it < 3; ++it) {
    int pt = wave + it*8;
    int mt = pt / 12, nt = pt % 12;
    int Mbase = mt*16, Nbase = nt*16;
    int ncol = Nbase + lr, bcol = ncol/12, tcol = ncol%12;
    int arow = (Mbase + lr) * CC;
    vf8 acc;
#pragma unroll
    for (int r = 0; r < 8; ++r) acc[r] = srb[Mbase + r + 8*half];
#pragma unroll
    for (int k0 = 0; k0 < CC; k0 += 4) {
      int kk = k0 + 2*half;
      vf2 a, bx;
      a.x  = sW[arow + kk];        a.y  = sW[arow + kk + 1];
      bx.x = sY[kk*NCOL + ncol];   bx.y = sY[(kk+1)*NCOL + ncol];
      acc = wmma_f32x4(a, bx, acc);
    }
    float s1 = 0.f, s2 = 0.f;
#pragma unroll
    for (int r = 0; r < 8; ++r) {
      int m = Mbase + r + 8*half;
      int gidx = ((bcol*CC + m)*NN + n)*TT + tcol;
      float z = acc[r] + X[gidx];     // residual add
      Y[gidx] = z;                    // post-residual, pre-norm
      s1 += z; s2 += z*z;
    }
    sPS[it*256 + tid] = s1;
    sPQ[it*256 + tid] = s2;
  }
  __syncthreads();

  // skip projection uses the PRE-residual gated x, still resident in LDS
  for (int idx = tid; idx < BB*CC; idx += 256) {
    int b = idx >> 5, c = idx & 31;
    float acc = skb[layer*CC + c];
    const float* wrow = &skw[(layer*CC + c)*TT];
    for (int t = 0; t < TT; ++t) acc += sY[c*NCOL + b*TT + t] * wrow[t];
    skip[(b*CC + c)*NN + n] += acc;   // unique writer per element
  }
  // deterministic per-batch layernorm partials (fixed-order scan, no atomics)
  if (tid < BB) {
    float a1 = 0.f, a2 = 0.f;
    for (int it = 0; it < 3; ++it)
      for (int t2 = 0; t2 < 256; ++t2) {
        int w2 = t2 >> 5, pt = w2 + it*8, nt = pt % 12;
        int b2 = (nt*16 + (t2 & 15)) / 12;
        if (b2 == tid) { a1 += sPS[it*256 + t2]; a2 += sPQ[it*256 + t2]; }
      }
    part[n*32 + tid]      = a1;
    part[n*32 + 16 + tid] = a2;
  }
}

// ---------------------------------------------------------------- LN stats reduce
__global__ void k_stats(const float* __restrict__ part,
                        float* __restrict__ mu, float* __restrict__ rstd) {
  int b = threadIdx.x;
  if (b >= BB) return;
  float s1 = 0.f, s2 = 0.f;
  for (int n = 0; n < NN; ++n) { s1 += part[n*32 + b]; s2 += part[n*32 + 16 + b]; }
  const float invM = 1.0f / (float)(CC*NN*TT);
  float m = s1 * invM;
  float v = s2 * invM - m*m;
  mu[b] = m;
  rstd[b] = rsqrtf(v + 1e-5f);
}

// ---------------------------------------------------------------- LN apply
__global__ __launch_bounds__(256) void k_norm(const float* __restrict__ Y,
    const float* __restrict__ lnw, const float* __restrict__ lnb,
    const float* __restrict__ mu, const float* __restrict__ rstd,
    float* __restrict__ X, int layer) {
  int idx = blockIdx.x * 256 + threadIdx.x;          // grid sized exactly XEL
  int t = idx % TT; int r1 = idx / TT;
  int n = r1 % NN;  int r2 = r1 / NN;
  int c = r2 % CC;  int b = r2 / CC;
  int lidx = ((layer*CC + c)*NN + n)*TT + t;
  X[idx] = (Y[idx] - mu[b]) * rstd[b] * lnw[lidx] + lnb[lidx];
}

// ---------------------------------------------------------------- head
__global__ __launch_bounds__(256) void k_final(const float* __restrict__ X,
    const float* __restrict__ skip,
    const float* __restrict__ sEw, const float* __restrict__ sEb,
    const float* __restrict__ e1w, const float* __restrict__ e1b,
    const float* __restrict__ e2w, const float* __restrict__ e2b,
    float* __restrict__ out) {
  __shared__ float sA[8][CC];
  __shared__ float sB2[8][CC];
  int wave = threadIdx.x >> 5, lane = threadIdx.x & 31;
  int w = blockIdx.x * 8 + wave;                 // grid = 4000 -> w < 32000 always
  int b = w / NN, n = w % NN;
  float acc = sEb[lane];
  for (int t = 0; t < TT; ++t) acc += X[((b*CC + lane)*NN + n)*TT + t] * sEw[lane*TT + t];
  acc += skip[(b*CC + lane)*NN + n];
  sA[wave][lane] = fmaxf(acc, 0.f);
  __syncthreads();
  float y1 = e1b[lane];
  for (int c = 0; c < CC; ++c) y1 += sA[wave][c] * e1w[lane*CC + c];
  sB2[wave][lane] = fmaxf(y1, 0.f);
  __syncthreads();
  if (lane == 0) {
    float o = e2b[0];
    for (int c = 0; c < CC; ++c) o += sB2[wave][c] * e2w[c];
    out[b*NN + n] = o;
  }
}

// ================================================================ launch
extern "C" void kernel_launch(void* const* d_in, const int* in_sizes, int n_in,
                              void* d_out, int out_size, void* d_ws, size_t ws_size,
                              hipStream_t stream) {
  const float* inp   = (const float*)d_in[0];
  const float* adj   = (const float*)d_in[1];
  const float* chars = (const float*)d_in[2];
  const float* Wcw = (const float*)d_in[3];  const float* bcw = (const float*)d_in[4];
  const float* Wcb = (const float*)d_in[5];  const float* bcb = (const float*)d_in[6];
  const float* Wgw = (const float*)d_in[7];  const float* bgw = (const float*)d_in[8];
  const float* Wgb = (const float*)d_in[9];  const float* bgb = (const float*)d_in[10];
  const float* Wrw = (const float*)d_in[11]; const float* brw = (const float*)d_in[12];
  const float* Wrb = (const float*)d_in[13]; const float* brb = (const float*)d_in[14];
  const float* stw = (const float*)d_in[15]; const float* stb = (const float*)d_in[16];
  const float* s0w = (const float*)d_in[17]; const float* s0b = (const float*)d_in[18];
  const float* skw = (const float*)d_in[19]; const float* skb = (const float*)d_in[20];
  const float* sEw = (const float*)d_in[21]; const float* sEb = (const float*)d_in[22];
  const float* lnw = (const float*)d_in[23]; const float* lnb = (const float*)d_in[24];
  const float* e1w = (const float*)d_in[25]; const float* e1b = (const float*)d_in[26];
  const float* e2w = (const float*)d_in[27]; const float* e2b = (const float*)d_in[28];

  float* ws    = (float*)d_ws;
  float* ch    = ws;                    // 8000
  float* charm = ws + 8000;             // 8000
  float* mu    = ws + 16000;            // 16
  float* rstd  = ws + 16016;            // 16
  float* part  = ws + 16384;            // 64000 (per-node LN partials)
  float* skip  = ws + 80384;            // SKEL
  float* X     = ws + 80384 + SKEL;     // XEL
  float* Y     = ws + 80384 + SKEL + XEL; // XEL

  k_charm<<<(NN + 255)/256, 256, 0, stream>>>(inp, chars, charm);
  k_ch<<<NN, 128, 0, stream>>>(adj, charm, ch);
  k_init<<<XEL/256, 256, 0, stream>>>(inp, stw, stb, X);
  k_skip0<<<SKEL/256, 256, 0, stream>>>(inp, s0w, s0b, skip);

  const size_t convSmem = (size_t)(2*CC*KTOT + CC*BB*XEXT + 2*CC) * sizeof(float); // ~104 KB
  for (int l = 0; l < LL; ++l) {
    k_conv<<<NN, 256, convSmem, stream>>>(ch, Wcw, bcw, Wcb, bcb,
                                          Wgw, bgw, Wgb, bgb, X, Y, l);
    k_resid<<<NN, 256, 0, stream>>>(ch, Wrw, brw, Wrb, brb, skw, skb,
                                    X, Y, skip, part, l);
    k_stats<<<1, 16, 0, stream>>>(part, mu, rstd);
    k_norm<<<XEL/256, 256, 0, stream>>>(Y, lnw, lnb, mu, rstd, X, l);
  }
  k_final<<<(BB*NN)/8, 256, 0, stream>>>(X, skip, sEw, sEb, e1w, e1b, e2w, e2b,
                                         (float*)d_out);
}